// HTensorSquare_40896678592924
// MI455X (gfx1250) — compile-verified
//
#include <hip/hip_runtime.h>

// ---------------------------------------------------------------------------
// e3nn-style tensor square (16x0e + 16x1o) -> 16x(0e + 1o + 1e + 2e), 'uvw'.
// All 5 paths reduce to D_pm[s,w] = sum_{uv<256} T_pm[s,uv] * W_p[uv,w];
// T rows are cheap bilinears of x built in VGPRs, the K=256 contraction is
// done with V_WMMA_F32_16X16X4_F32 (fp32 matrix cores, 2048 FLOP/instr).
// One wave = one 16-sample tile; 13 fp32 16x16 accumulators per wave.
// Weights (5 x 16KB) are DMA'd into LDS with GLOBAL_LOAD_ASYNC_TO_LDS_B128
// (ASYNCcnt-tracked, no VGPR round-trip); x tile is repacked to float4
// (x0, x1.xyz) in LDS so each A fragment needs one ds_load_b128, and B
// operand pairs come from LDS as single ds_load_2addr_b32 ops.
// ---------------------------------------------------------------------------

typedef float v2f __attribute__((ext_vector_type(2)));
typedef float v8f __attribute__((ext_vector_type(8)));

#define WPG 8  // waves (16-sample tiles) per workgroup

__device__ __forceinline__ v8f wmma4(v2f a, v2f b, v8f c) {
  // D = A(16x4 f32) * B(4x16 f32) + C(16x16 f32)
  return __builtin_amdgcn_wmma_f32_16x16x4_f32(false, a, false, b, (short)0, c,
                                               false, false);
}

// 13 T-channel values for one (u,v) pair of one sample.
// xu/xv = (x0[u], x1[u].x, x1[u].y, x1[u].z)
__device__ __forceinline__ void make_t(const float4 xu, const float4 xv,
                                       float t[13]) {
  const float S2 = 0.70710678118654752f;  // 1/sqrt(2)
  const float S3 = 0.57735026918962576f;  // 1/sqrt(3)
  const float S6 = 0.40824829046386302f;  // 1/sqrt(6)
  t[0] = xu.x * xv.x;                                     // 0e x 0e -> 0e
  t[1] = S3 * (xu.y * xv.y + xu.z * xv.z + xu.w * xv.w);  // 1o . 1o -> 0e
  t[2] = xu.x * xv.y;                                     // 0e x 1o -> 1o
  t[3] = xu.x * xv.z;
  t[4] = xu.x * xv.w;
  t[5] = S2 * (xu.z * xv.w - xu.w * xv.z);                // 1o x 1o -> 1e
  t[6] = S2 * (xu.w * xv.y - xu.y * xv.w);
  t[7] = S2 * (xu.y * xv.z - xu.z * xv.y);
  t[8] = S2 * (xu.y * xv.z + xu.z * xv.y);                // 1o x 1o -> 2e
  t[9] = S2 * (xu.z * xv.w + xu.w * xv.z);
  t[10] = S6 * (2.0f * xu.w * xv.w - xu.y * xv.y - xu.z * xv.z);
  t[11] = S2 * (xu.y * xv.w + xu.w * xv.y);
  t[12] = S2 * (xu.y * xv.y - xu.z * xv.z);
}

__global__ __launch_bounds__(32 * WPG) void tsq_wmma_kernel(
    const float* __restrict__ x, const float* __restrict__ w0,
    const float* __restrict__ w1, const float* __restrict__ w2,
    const float* __restrict__ w3, const float* __restrict__ w4,
    float* __restrict__ out, int nsamp, int nblk) {
  extern __shared__ float smem[];
  float* sW = smem;                         // 5 * 4096 floats (W_p[uv*16+w])
  float4* sX = (float4*)(smem + 5 * 4096);  // WPG tiles * 16 samples * 16 u

  const int tid = threadIdx.x;

  {  // Async-DMA all five uvw weight tensors straight into LDS.
     // GVS mode: per-lane LDS dest VGPR + 32-bit global offset VGPR + SGPR
     // base pair. Tracked by ASYNCcnt (no VGPR round-trip, no ds_store).
    const float* ws[5] = {w0, w1, w2, w3, w4};
#pragma unroll
    for (int p = 0; p < 5; ++p) {
      const float* src = ws[p];
      for (int i = tid; i < 1024; i += 32 * WPG) {
        unsigned lds_addr = (unsigned)(uintptr_t)(sW + p * 4096 + i * 4);
        unsigned goff = (unsigned)(i * 16);  // byte offset from src
        asm volatile("global_load_async_to_lds_b128 %0, %1, %2"
                     :
                     : "v"(lds_addr), "v"(goff), "s"(src)
                     : "memory");
      }
    }
  }

  const int wave = tid >> 5;
  const int lane = tid & 31;
  const int blk = blockIdx.x * WPG + wave;
  float4* xs = sX + wave * 256;  // [sample][u] = (x0[u], x1[u].xyz)

  if (blk < nblk) {
    for (int e = lane; e < 256; e += 32) {
      const int s = e >> 4, u = e & 15;
      int z = blk * 16 + s;
      if (z >= nsamp) z = nsamp - 1;  // clamp tail tiles (loads only)
      const float* xp = x + (size_t)z * 64;
      float4 v;
      v.x = xp[u];
      v.y = xp[16 + 3 * u + 0];
      v.z = xp[16 + 3 * u + 1];
      v.w = xp[16 + 3 * u + 2];
      xs[s * 16 + u] = v;
    }
    if (blk + WPG < nblk)  // pull next tile's x toward the caches
      __builtin_prefetch(x + (size_t)(blk + WPG) * 16 * 64 + lane * 32, 0, 1);
  }
  // All async weight transfers must land in LDS before any wave reads them.
  asm volatile("s_wait_asynccnt 0x0" ::: "memory");
  __syncthreads();  // every thread reaches this (no early returns above)
  if (blk >= nblk) return;

  const int h = lane >> 4;  // K-half inside a K=4 WMMA chunk
  const int r = lane & 15;  // A row (= sample) and B column (= output w)

  v8f c000 = {}, c110 = {};
  v8f c011[3] = {{}, {}, {}};
  v8f c111[3] = {{}, {}, {}};
  v8f c112[5] = {{}, {}, {}, {}, {}};

  for (int u = 0; u < 16; ++u) {
    const float4 xu = xs[r * 16 + u];
#pragma unroll
    for (int cc = 0; cc < 4; ++cc) {
      const int v0 = cc * 4 + 2 * h;  // v index (uv mod 16) for A/B slot 0
      const float4 xva = xs[r * 16 + v0];
      const float4 xvb = xs[r * 16 + v0 + 1];
      float ta[13], tb[13];
      make_t(xu, xva, ta);
      make_t(xu, xvb, tb);
      v2f a[13];
#pragma unroll
      for (int q = 0; q < 13; ++q) {
        a[q].x = ta[q];
        a[q].y = tb[q];
      }
      const int row = (u * 16 + v0) * 16 + r;  // W_p[uv, w=r]
      v2f b[5];
#pragma unroll
      for (int p = 0; p < 5; ++p) {
        b[p].x = sW[p * 4096 + row];
        b[p].y = sW[p * 4096 + row + 16];
      }
      c000 = wmma4(a[0], b[0], c000);
      c110 = wmma4(a[1], b[2], c110);
      c011[0] = wmma4(a[2], b[1], c011[0]);
      c011[1] = wmma4(a[3], b[1], c011[1]);
      c011[2] = wmma4(a[4], b[1], c011[2]);
      c111[0] = wmma4(a[5], b[3], c111[0]);
      c111[1] = wmma4(a[6], b[3], c111[1]);
      c111[2] = wmma4(a[7], b[3], c111[2]);
      c112[0] = wmma4(a[8], b[4], c112[0]);
      c112[1] = wmma4(a[9], b[4], c112[1]);
      c112[2] = wmma4(a[10], b[4], c112[2]);
      c112[3] = wmma4(a[11], b[4], c112[3]);
      c112[4] = wmma4(a[12], b[4], c112[4]);
    }
  }

  const float A1 = 0.0625f;         // 1/sqrt(16*16)   (single-path outputs)
  const float A2 = 0.04419417382f;  // 1/sqrt(16*16*2) (0e gets 2 paths)
#pragma unroll
  for (int j = 0; j < 8; ++j) {
    const int z = blk * 16 + 8 * h + j;  // D row = sample
    if (z >= nsamp) continue;
    float* o = out + (size_t)z * 192;
    o[r] = A2 * (c000[j] + c110[j]);  // 16x0e
#pragma unroll
    for (int m = 0; m < 3; ++m) o[16 + r * 3 + m] = A1 * c011[m][j];  // 16x1o
#pragma unroll
    for (int m = 0; m < 3; ++m) o[64 + r * 3 + m] = A1 * c111[m][j];  // 16x1e
#pragma unroll
    for (int m = 0; m < 5; ++m) o[112 + r * 5 + m] = A1 * c112[m][j];  // 16x2e
  }
}

extern "C" void kernel_launch(void* const* d_in, const int* in_sizes, int n_in,
                              void* d_out, int out_size, void* d_ws,
                              size_t ws_size, hipStream_t stream) {
  const float* x = (const float*)d_in[0];
  const float* w0 = (const float*)d_in[1];
  const float* w1 = (const float*)d_in[2];
  const float* w2 = (const float*)d_in[3];
  const float* w3 = (const float*)d_in[4];
  const float* w4 = (const float*)d_in[5];
  float* out = (float*)d_out;

  const int nsamp = in_sizes[0] / 64;        // 100000
  const int nblk = (nsamp + 15) / 16;        // 6250 16-sample tiles
  const int ngrid = (nblk + WPG - 1) / WPG;  // 782 workgroups

  const size_t shmem = (size_t)(5 * 4096 + WPG * 256 * 4) * sizeof(float);

  tsq_wmma_kernel<<<dim3(ngrid), dim3(32 * WPG), shmem, stream>>>(
      x, w0, w1, w2, w3, w4, out, nsamp, nblk);
}